// SubGroupPooler_59708635349141
// MI455X (gfx1250) — compile-verified
//
#include <hip/hip_runtime.h>

// SubGroupPooler: Z[t, s*D + d] = max_{g} P[idx[s][g], t, d]
// Streaming 6-way elementwise max; memory-bound (403 MB read + 67 MB write
// -> ~20 us floor at 23.3 TB/s). No contraction dim => WMMA inapplicable.
// Strategy: 24 batched b128 non-temporal loads per thread (one clause, max
// MLP), shared base addresses + immediate offsets, constant-folded store
// offsets, zero per-k address math, no bounds checks (exact tiling).

typedef __attribute__((ext_vector_type(4))) float v4f;

namespace {
constexpr int N_SUB  = 2;     // subgroups (teams)
constexpr int GSIZE  = 6;     // players per subgroup
constexpr int T_DIM  = 4096;  // frames
constexpr int D_DIM  = 2048;  // fused embed dim
constexpr int D4     = D_DIM / 4;                        // float4 per row = 512
constexpr long long PLANE  = (long long)T_DIM * D_DIM;   // elems per player
constexpr long long PLANE4 = PLANE / 4;                  // float4 per player
constexpr int BLOCK  = 256;   // 8 wave32 waves
constexpr int UNROLL = 4;     // float4 per thread
constexpr long long TILE4 = (long long)BLOCK * UNROLL;   // float4 per block = 1024

static_assert(PLANE4 % TILE4 == 0, "grid must tile the plane exactly");
static_assert(BLOCK <= D4, "store-offset folding needs tid < D4");
static_assert((TILE4 % D4) == 0, "block tile must be row-aligned");
}

__global__ __launch_bounds__(BLOCK)
void SubGroupPooler_maxpool_kernel(const float* __restrict__ P,
                                   const int*   __restrict__ idx,
                                   float*       __restrict__ Z)
{
    const int s = blockIdx.y;                 // subgroup id, 0..N_SUB-1
    const long long flat0 = (long long)blockIdx.x * TILE4 + threadIdx.x;

    // One base address per input stream; the k offsets are immediates.
    const v4f* __restrict__ base[GSIZE];
#pragma unroll
    for (int g = 0; g < GSIZE; ++g) {
        const long long n = (long long)idx[s * GSIZE + g];   // scalarized load
        base[g] = reinterpret_cast<const v4f*>(P + n * PLANE) + flat0;
    }

    // Batch all 24 b128 NT loads up front: one load clause, 24 in flight.
    v4f v[GSIZE][UNROLL];
#pragma unroll
    for (int g = 0; g < GSIZE; ++g)
#pragma unroll
        for (int k = 0; k < UNROLL; ++k)
            v[g][k] = __builtin_nontemporal_load(base[g] + k * BLOCK);

    // Output base: flat0 = t0*D4 + d40 with d40 = tid < D4 (no wrap inside a
    // k-pair); out index = t*(N_SUB*D4) + s*D4 + d4.
    const long long t0  = flat0 >> 9;          // / D4
    const long long d40 = flat0 & (D4 - 1);    // % D4
    v4f* __restrict__ outp = reinterpret_cast<v4f*>(Z)
        + t0 * (N_SUB * (long long)D4) + (long long)s * D4 + d40;

#pragma unroll
    for (int k = 0; k < UNROLL; ++k) {
        v4f m = v[0][k];
#pragma unroll
        for (int g = 1; g < GSIZE; ++g)
            m = __builtin_elementwise_max(m, v[g][k]);

        // Constant-folded store offset: rows advance by N_SUB*D4 in out space.
        const int kk   = k * BLOCK;
        const int koff = (kk / D4) * (N_SUB * D4) + (kk % D4);  // {0,256,1024,1280}
        __builtin_nontemporal_store(m, outp + koff);
    }
}

extern "C" void kernel_launch(void* const* d_in, const int* in_sizes, int n_in,
                              void* d_out, int out_size, void* d_ws, size_t ws_size,
                              hipStream_t stream)
{
    const float* P   = (const float*)d_in[0];   // [12, 4096, 2048] f32
    const int*   idx = (const int*)d_in[1];     // [2, 6]
    float*       Z   = (float*)d_out;           // [1, 4096, 4096] f32

    dim3 grid((unsigned)(PLANE4 / TILE4), N_SUB, 1);  // 2048 x 2
    dim3 block(BLOCK, 1, 1);

    SubGroupPooler_maxpool_kernel<<<grid, block, 0, stream>>>(P, idx, Z);
}